// GUIElementDetector_61091614818665
// MI455X (gfx1250) — compile-verified
//
#include <hip/hip_runtime.h>
#include <hip/hip_bf16.h>
#include <stdint.h>

// ---------------------------------------------------------------------------
// GUI element detector feature extractor (color hist + LBP hist + geom)
//   input : img (128, 320, 320, 3) fp32 in [0,256)
//   output: (128, 512) fp32 = [color(48) | lbp(256) | geom(4) | zeros(204)]
//
// MI455X reasoning: 157 MB streamed once -> ~6.7 us HBM floor (23.3 TB/s).
// Memory-bound histogram; WMMA not applicable (no matmul structure; a one-hot
// matmul histogram would inflate work x256). CDNA5 paths used instead:
//   - async global->LDS staging (global_load_async_to_lds_b128 +
//     s_wait_asynccnt) for VGPR-free row streaming overlapped with LDS init
//   - wave32-private LDS histograms (8 replicas) to cut ds_add_u32 contention
//   - whole input fits in the 192 MB L2, so the 2-row halo re-read is free
// ---------------------------------------------------------------------------

#define B_IMG   128
#define H_IMG   320
#define W_IMG   320
#define R_TILE  8                      // image rows per block
#define ROWS_ST (R_TILE + 2)           // staged rows incl. halo = 10
#define TPI     (H_IMG / R_TILE)       // tiles per image = 40
#define ROWF    (W_IMG * 3)            // floats per row = 960
#define ROWBYTES (ROWF * 4)            // 3840 bytes per row
#define CHUNKS  (ROWBYTES / 16)        // 240 b128 chunks per row
#define NTHREADS 256                   // 8 wave32
#define NWAVES   8
#define FEAT_COLOR 48
#define FEAT_LBP   256
#define OUT_LEN    512
#define GRAY_W  (W_IMG + 2)            // 322

// Flat pointer to a __shared__ object: low 32 bits are the LDS byte offset
// (amdgcn generic shared address = {shared aperture hi32, lds offset}).
__device__ __forceinline__ unsigned lds_off(const void* p) {
    return (unsigned)(unsigned long long)(uintptr_t)p;
}

__device__ __forceinline__ void wait_asynccnt0() {
#if __has_builtin(__builtin_amdgcn_s_wait_asynccnt)
    __builtin_amdgcn_s_wait_asynccnt(0);
#else
    asm volatile("s_wait_asynccnt 0" ::: "memory");
#endif
}

__global__ void __launch_bounds__(NTHREADS)
hist_kernel(const float* __restrict__ img,
            unsigned* __restrict__ hcolor_g,   // [B][48]
            unsigned* __restrict__ hlbp_g) {   // [B][256]
    __shared__ float    raw[ROWS_ST * ROWF];            // 38400 B staged pixels
    __shared__ int      gray[ROWS_ST * GRAY_W];         // 12880 B gray + col halo
    __shared__ unsigned hc[NWAVES][FEAT_COLOR];         //  1536 B
    __shared__ unsigned hl[NWAVES][FEAT_LBP];           //  8192 B  (total ~61 KB)

    const int tid    = threadIdx.x;
    const int wid    = tid >> 5;                        // wave32 id 0..7
    const int blk    = blockIdx.x;
    const int imgIdx = blk / TPI;
    const int tile   = blk - imgIdx * TPI;
    const int r0     = tile * R_TILE;

    // ---- Phase A: async-stage 10 rows (edge-clamped halo) into LDS --------
    // One b128 async load per lane per row: 240 lanes x 10 rows, row index
    // uniform (scalar loop), so address math is cheap.
    {
        const unsigned rawBase = lds_off(&raw[0]);
        const char* base = (const char*)img + (size_t)imgIdx * H_IMG * ROWBYTES;
#pragma unroll
        for (int row = 0; row < ROWS_ST; ++row) {
            int grow = r0 - 1 + row;
            grow = grow < 0 ? 0 : (grow > H_IMG - 1 ? H_IMG - 1 : grow);
            if (tid < CHUNKS) {
                unsigned long long gaddr = (unsigned long long)(uintptr_t)
                    (base + (size_t)grow * ROWBYTES + (size_t)(tid * 16));
                unsigned laddr = rawBase + (unsigned)(row * ROWBYTES + tid * 16);
                asm volatile("global_load_async_to_lds_b128 %0, %1, off"
                             :: "v"(laddr), "v"(gaddr) : "memory");
            }
        }
    }

    // ---- zero wave-private histograms while async loads are in flight -----
    for (int i = tid; i < NWAVES * FEAT_COLOR; i += NTHREADS) (&hc[0][0])[i] = 0u;
    for (int i = tid; i < NWAVES * FEAT_LBP;   i += NTHREADS) (&hl[0][0])[i] = 0u;

    wait_asynccnt0();
    __syncthreads();

    // ---- Phase B: grayscale tile (with clamped column halo) + color hist --
#pragma unroll 2
    for (int i = 0; i < ROWS_ST; ++i) {
        const float* rowp = &raw[i * ROWF];
        int* grow_p = &gray[i * GRAY_W];
        for (int cc = tid; cc < GRAY_W; cc += NTHREADS) {
            int col = cc - 1;
            col = col < 0 ? 0 : (col > W_IMG - 1 ? W_IMG - 1 : col);
            const float* p = rowp + col * 3;
            float g = 0.114f * p[0] + 0.587f * p[1] + 0.299f * p[2];
            int gi = (int)rintf(g);                     // jnp.round == RNE
            gi = gi < 0 ? 0 : (gi > 255 ? 255 : gi);
            grow_p[cc] = gi;
        }
    }

#pragma unroll 2
    for (int i = 1; i <= R_TILE; ++i) {                 // interior rows only
        const float* rowp = &raw[i * ROWF];
        for (int c = tid; c < W_IMG; c += NTHREADS) {
            const float* p = rowp + c * 3;
#pragma unroll
            for (int ch = 0; ch < 3; ++ch) {
                int pix = (int)p[ch];                   // trunc == astype(int32)
                pix = pix < 0 ? 0 : (pix > 255 ? 255 : pix);
                atomicAdd(&hc[wid][ch * 16 + (pix >> 4)], 1u);
            }
        }
    }
    __syncthreads();

    // ---- Phase C: LBP over interior pixels --------------------------------
#pragma unroll 2
    for (int i = 1; i <= R_TILE; ++i) {
        const int* grow_p = &gray[i * GRAY_W];
        for (int c = tid; c < W_IMG; c += NTHREADS) {
            const int* g = grow_p + 1 + c;
            const int S = GRAY_W;
            int center = g[0];
            // offsets in reference order:
            // (-1,-1)(-1,0)(-1,1)(0,1)(1,1)(1,0)(1,-1)(0,-1)
            int pat = 0;
            pat |= (g[-S - 1] >= center) << 0;
            pat |= (g[-S    ] >= center) << 1;
            pat |= (g[-S + 1] >= center) << 2;
            pat |= (g[ 1    ] >= center) << 3;
            pat |= (g[ S + 1] >= center) << 4;
            pat |= (g[ S    ] >= center) << 5;
            pat |= (g[ S - 1] >= center) << 6;
            pat |= (g[-1    ] >= center) << 7;
            atomicAdd(&hl[wid][pat], 1u);
        }
    }
    __syncthreads();

    // ---- Phase D: flush wave-private histograms to global counters --------
    if (tid < FEAT_COLOR) {
        unsigned s = 0;
#pragma unroll
        for (int w = 0; w < NWAVES; ++w) s += hc[w][tid];
        atomicAdd(&hcolor_g[imgIdx * FEAT_COLOR + tid], s);
    }
    {
        unsigned s = 0;
#pragma unroll
        for (int w = 0; w < NWAVES; ++w) s += hl[w][tid];
        atomicAdd(&hlbp_g[imgIdx * FEAT_LBP + tid], s);
    }
}

__global__ void __launch_bounds__(NTHREADS)
zero_ws_kernel(unsigned* __restrict__ ws, int n) {
    int i = blockIdx.x * NTHREADS + threadIdx.x;
    if (i < n) ws[i] = 0u;
}

__global__ void __launch_bounds__(NTHREADS)
finalize_kernel(const unsigned* __restrict__ hcolor_g,
                const unsigned* __restrict__ hlbp_g,
                float* __restrict__ out) {
    __shared__ float red[NTHREADS];
    const int img = blockIdx.x;
    const int t   = threadIdx.x;

    float lv = (float)hlbp_g[img * FEAT_LBP + t];
    red[t] = lv;
    __syncthreads();
    for (int s = NTHREADS / 2; s > 0; s >>= 1) {
        if (t < s) red[t] += red[t + s];
        __syncthreads();
    }
    float lsum = red[0];
    __syncthreads();

    float cv = (t < FEAT_COLOR) ? (float)hcolor_g[img * FEAT_COLOR + t] : 0.0f;
    red[t] = cv;
    __syncthreads();
    for (int s = NTHREADS / 2; s > 0; s >>= 1) {
        if (t < s) red[t] += red[t + s];
        __syncthreads();
    }
    float csum = red[0];

    float* o = out + (size_t)img * OUT_LEN;
    if (t < FEAT_COLOR)            o[t]      = cv / (csum + 1e-6f);
    /* all 256 threads */          o[48 + t] = lv / (lsum + 1e-6f);

    // geom = [W/H, W*H, W, H] / 1000 at 304..307, zeros through 511
    for (int j = 304 + t; j < OUT_LEN; j += NTHREADS) {
        float g = 0.0f;
        if (j == 304) g = 0.001f;          // (320/320)/1000
        else if (j == 305) g = 102.4f;     // 320*320/1000
        else if (j == 306) g = 0.32f;      // 320/1000
        else if (j == 307) g = 0.32f;      // 320/1000
        o[j] = g;
    }
}

extern "C" void kernel_launch(void* const* d_in, const int* in_sizes, int n_in,
                              void* d_out, int out_size, void* d_ws, size_t ws_size,
                              hipStream_t stream) {
    const float* img = (const float*)d_in[0];
    unsigned* hc = (unsigned*)d_ws;                       // [128][48]
    unsigned* hl = hc + B_IMG * FEAT_COLOR;               // [128][256]
    float* out = (float*)d_out;

    const int nz = B_IMG * (FEAT_COLOR + FEAT_LBP);       // 38912 uints (~152 KB ws)
    zero_ws_kernel<<<(nz + NTHREADS - 1) / NTHREADS, NTHREADS, 0, stream>>>(hc, nz);
    hist_kernel<<<B_IMG * TPI, NTHREADS, 0, stream>>>(img, hc, hl);
    finalize_kernel<<<B_IMG, NTHREADS, 0, stream>>>(hc, hl, out);
}